// SpectralConv2d_7112465842486
// MI455X (gfx1250) — compile-verified
//
#include <hip/hip_runtime.h>

typedef __attribute__((ext_vector_type(2))) float v2f;
typedef __attribute__((ext_vector_type(8))) float v8f;

#define PI2 6.28318530717958647692f

// Basis table geometry (K-pair interleaved, bank-staggered row strides).
#define FWP_STRIDE 96     // pair-row stride (floats); 96 % 64 == 32 -> half-waves on disjoint banks
#define FWP_SIZE   (128 * FWP_STRIDE)   // 12288 floats
#define CWP_STRIDE 544    // 512 + 32
#define CWP_SIZE   (16 * CWP_STRIDE)    // 8704 floats

// ---------------------------------------------------------------------------
// One-shot table build: Fw (forward w-DFT basis), Cw (inverse w-rDFT basis
// with Hermitian 2x folded in), tw (cos/sin of 2*pi*t/256).
// fwp[(w>>1)*96  + k*2 + (w&1)], k = 2*kx + (0=re,1=im), 256 w x 32 k
// cwp[(k>>1)*544 + w*2 + (k&1)], 32 k x 256 w
// ---------------------------------------------------------------------------
__global__ __launch_bounds__(256) void fno_build_tables(
    float* __restrict__ fwp, float* __restrict__ cwp, float* __restrict__ tw) {
  const int t = threadIdx.x;
  for (int e = t; e < 8192; e += 256) {
    int w = e >> 5, k = e & 31, kx = k >> 1;
    float s, c;
    __sincosf(PI2 * (float)((w * kx) & 255) * (1.0f / 256.0f), &s, &c);
    fwp[(w >> 1) * FWP_STRIDE + k * 2 + (w & 1)] =
        (k & 1) ? (-s * (1.0f / 16.0f)) : (c * (1.0f / 16.0f));
  }
  for (int e = t; e < 8192; e += 256) {
    int k = e >> 8, w = e & 255, kx = k >> 1;
    float s, c;
    __sincosf(PI2 * (float)((w * kx) & 255) * (1.0f / 256.0f), &s, &c);
    float v;
    if (kx == 0)
      v = (k & 1) ? 0.0f : (1.0f / 16.0f);
    else
      v = (k & 1) ? (-2.0f * s * (1.0f / 16.0f)) : (2.0f * c * (1.0f / 16.0f));
    cwp[(k >> 1) * CWP_STRIDE + w * 2 + (k & 1)] = v;
  }
  {
    float s, c;
    __sincosf(PI2 * (float)t * (1.0f / 256.0f), &s, &c);
    tw[t * 2] = c;
    tw[t * 2 + 1] = s;
  }
}

// ---------------------------------------------------------------------------
// Stage A: partial forward DFT over w (16 complex modes), f32 WMMA GEMM.
// x[bc][h][w] -> X1[bc][h][k32];  M=rows(b,c,h) K=256(w) N=32
// ---------------------------------------------------------------------------
__global__ __launch_bounds__(128) void fno_dft_w(const float* __restrict__ x,
                                                 const float* __restrict__ fwp,
                                                 float* __restrict__ X1) {
  __shared__ float xs[64 * 260];       // 64 rows, pad -> conflict-free A frags
  __shared__ float fwl[FWP_SIZE];      // paired basis
  const int t = threadIdx.x;
  const int lane = t & 31;
  const int wave = t >> 5;

  // Copy basis table (L2-hot) into LDS.
  for (int i = 0; i < FWP_SIZE / 4 / 128; ++i) {
    int f = t + i * 128;
    reinterpret_cast<float4*>(fwl)[f] = reinterpret_cast<const float4*>(fwp)[f];
  }
  const int bc = blockIdx.x >> 2;            // image index (b*32+c)
  const int hblk = (blockIdx.x & 3) * 64;    // 64 h-rows per block
  const float* xsrc = x + (size_t)bc * 65536 + (size_t)hblk * 256;
  for (int i = 0; i < 32; ++i) {
    int f = t + i * 128;                     // float4 id, 4096 total
    int row = f >> 6, c4 = f & 63;
    const float4 v = reinterpret_cast<const float4*>(xsrc)[f];
    *reinterpret_cast<float4*>(&xs[row * 260 + c4 * 4]) = v;
  }
  __syncthreads();

  // WMMA f32 16x16x4: lanes 0-15 hold K=k0,k0+1; lanes 16-31 hold K=k0+2,k0+3
  const int mrow = lane & 15;
  const int khalf = lane >> 4;               // 0 or 1 (K offset /2)
  const float* arow = &xs[(wave * 16 + mrow) * 260 + khalf * 2];
  const float* brow = &fwl[khalf * FWP_STRIDE + mrow * 2];
  v8f c0 = {};
  v8f c1 = {};
  for (int kk = 0; kk < 64; ++kk) {
    v2f a = *reinterpret_cast<const v2f*>(&arow[kk * 4]);
    v2f b0 = *reinterpret_cast<const v2f*>(&brow[kk * 2 * FWP_STRIDE]);
    v2f b1 = *reinterpret_cast<const v2f*>(&brow[kk * 2 * FWP_STRIDE + 32]);
    c0 = __builtin_amdgcn_wmma_f32_16x16x4_f32(false, a, false, b0, (short)0,
                                               c0, false, false);
    c1 = __builtin_amdgcn_wmma_f32_16x16x4_f32(false, a, false, b1, (short)0,
                                               c1, false, false);
  }

  // Store C tiles: lane = column k, VGPR j = row h (lanes>=16 -> +8).
  float* dst = X1 + (size_t)bc * 8192;
  const int hb = hblk + wave * 16 + (khalf << 3);
  for (int j = 0; j < 8; ++j) {
    dst[(hb + j) * 32 + mrow] = c0[j];
    dst[(hb + j) * 32 + 16 + mrow] = c1[j];
  }
}

// ---------------------------------------------------------------------------
// Stage B: forward DFT over h, keep 16 modes.
// X1[bc][h][k32] -> X2[bc][m=ky*16+kx][2]
// ---------------------------------------------------------------------------
__global__ __launch_bounds__(256) void fno_dft_h(const float* __restrict__ X1,
                                                 const float* __restrict__ tw,
                                                 float* __restrict__ X2) {
  __shared__ float x1s[8192];
  __shared__ float tws[512];
  const int t = threadIdx.x;
  if (t < 128)
    reinterpret_cast<float4*>(tws)[t] = reinterpret_cast<const float4*>(tw)[t];
  const float* src = X1 + (size_t)blockIdx.x * 8192;
  for (int i = 0; i < 8; ++i) {
    int f = t + i * 256;
    reinterpret_cast<float4*>(x1s)[f] = reinterpret_cast<const float4*>(src)[f];
  }
  __syncthreads();

  const int ky = t >> 4, kx = t & 15;
  float are = 0.f, aim = 0.f;
  for (int h = 0; h < 256; ++h) {
    float a = x1s[h * 32 + 2 * kx];
    float b = x1s[h * 32 + 2 * kx + 1];
    int idx = (ky * h) & 255;
    float c = tws[idx * 2], s = tws[idx * 2 + 1];
    are += a * c + b * s;   // (a+ib)(c-is)
    aim += b * c - a * s;
  }
  float* dst = X2 + (size_t)blockIdx.x * 512;
  dst[t * 2] = are * (1.0f / 16.0f);
  dst[t * 2 + 1] = aim * (1.0f / 16.0f);
}

// ---------------------------------------------------------------------------
// Stage C+D1: complex channel mix then inverse DFT over h.
// X2[b][i][m][2], wgt[i][o][m][2] -> Y1 (pair layout [bo][kx][h][2])
// ---------------------------------------------------------------------------
__global__ __launch_bounds__(256) void fno_mix_idft_h(
    const float* __restrict__ X2, const float* __restrict__ wgt,
    const float* __restrict__ tw, float* __restrict__ Y1) {
  __shared__ float x2s[16384];
  __shared__ float wls[16384];
  __shared__ float mo[512];
  __shared__ float tws[512];
  const int t = threadIdx.x;
  const int b = blockIdx.x >> 5;
  const int o = blockIdx.x & 31;
  if (t < 128)
    reinterpret_cast<float4*>(tws)[t] = reinterpret_cast<const float4*>(tw)[t];
  const float* xsrc = X2 + (size_t)b * 16384;
  for (int i = 0; i < 16; ++i) {
    int f = t + i * 256;
    reinterpret_cast<float4*>(x2s)[f] =
        reinterpret_cast<const float4*>(xsrc)[f];
  }
  for (int i = 0; i < 32; ++i) {
    const float* wsrc = wgt + ((size_t)(i * 32 + o)) * 512;
    wls[i * 512 + t] = wsrc[t];
    wls[i * 512 + 256 + t] = wsrc[256 + t];
  }
  __syncthreads();

  {
    float re = 0.f, im = 0.f;
    for (int i = 0; i < 32; ++i) {
      float xr = x2s[i * 512 + 2 * t], xi = x2s[i * 512 + 2 * t + 1];
      float wr = wls[i * 512 + 2 * t], wi = wls[i * 512 + 2 * t + 1];
      re += xr * wr - xi * wi;
      im += xr * wi + xi * wr;
    }
    mo[2 * t] = re;
    mo[2 * t + 1] = im;
  }
  __syncthreads();

  // Inverse DFT over h; write re/im adjacent (pair layout) as float2.
  const int h = t;
  float* dst = Y1 + (size_t)blockIdx.x * 8192;
  for (int kx = 0; kx < 16; ++kx) {
    float re = 0.f, im = 0.f;
    for (int ky = 0; ky < 16; ++ky) {
      float mr = mo[(ky * 16 + kx) * 2], mi = mo[(ky * 16 + kx) * 2 + 1];
      int idx = (ky * h) & 255;
      float c = tws[idx * 2], s = tws[idx * 2 + 1];
      re += mr * c - mi * s;   // (mr+i mi)(c+is)
      im += mr * s + mi * c;
    }
    float2 v = make_float2(re * (1.0f / 16.0f), im * (1.0f / 16.0f));
    *reinterpret_cast<float2*>(&dst[kx * 512 + h * 2]) = v;
  }
}

// ---------------------------------------------------------------------------
// Stage D2: inverse rDFT over w, f32 WMMA GEMM: M=rows(b,o,h) K=32 N=256.
// Y1 pair layout [bo][kx][h][2] -> out[b][o][h][w]
// ---------------------------------------------------------------------------
__global__ __launch_bounds__(256) void fno_idft_w(const float* __restrict__ Y1,
                                                  const float* __restrict__ cwp,
                                                  float* __restrict__ out) {
  __shared__ float cwl[CWP_SIZE];
  const int t = threadIdx.x;
  const int lane = t & 31;
  const int wave = t >> 5;
  for (int e = t; e < CWP_SIZE / 4; e += 256)
    reinterpret_cast<float4*>(cwl)[e] = reinterpret_cast<const float4*>(cwp)[e];
  __syncthreads();

  const int bo = blockIdx.x >> 1;
  const int hb = (blockIdx.x & 1) * 128 + wave * 16;
  const int mrow = lane & 15;
  const int khalf = lane >> 4;

  // Preload A fragments: contiguous 8B per lane (128B per half-wave).
  const float* ysrc = Y1 + (size_t)bo * 8192 + khalf * 512 + (hb + mrow) * 2;
  v2f afr[8];
  for (int kk = 0; kk < 8; ++kk)
    afr[kk] = *reinterpret_cast<const v2f*>(&ysrc[kk * 2 * 512]);

  float* dst = out + (size_t)bo * 65536;
  const int hrow = hb + (khalf << 3);
  const float* brow = &cwl[khalf * CWP_STRIDE + mrow * 2];
  for (int wt = 0; wt < 16; ++wt) {   // 16 column tiles cover w=0..255
    v8f acc = {};
    for (int kk = 0; kk < 8; ++kk) {
      v2f b = *reinterpret_cast<const v2f*>(
          &brow[kk * 2 * CWP_STRIDE + wt * 32]);
      acc = __builtin_amdgcn_wmma_f32_16x16x4_f32(false, afr[kk], false, b,
                                                  (short)0, acc, false, false);
    }
    for (int j = 0; j < 8; ++j)
      dst[(hrow + j) * 256 + wt * 16 + mrow] = acc[j];
  }
}

extern "C" void kernel_launch(void* const* d_in, const int* in_sizes, int n_in,
                              void* d_out, int out_size, void* d_ws,
                              size_t ws_size, hipStream_t stream) {
  (void)in_sizes; (void)n_in; (void)out_size; (void)ws_size;
  const float* x = (const float*)d_in[0];     // (16,32,256,256) f32
  const float* wgt = (const float*)d_in[1];   // (32,32,16,16,2) f32
  float* out = (float*)d_out;                 // (16,32,256,256) f32

  float* X1 = (float*)d_ws;                   // 4,194,304 floats
  float* Y1 = X1 + 4194304;                   // 4,194,304 floats
  float* X2 = Y1 + 4194304;                   //   262,144 floats
  float* fwp = X2 + 262144;                   // FWP_SIZE floats
  float* cwp = fwp + FWP_SIZE;                // CWP_SIZE floats
  float* tw = cwp + CWP_SIZE;                 // 512 floats

  fno_build_tables<<<1, 256, 0, stream>>>(fwp, cwp, tw);
  fno_dft_w<<<2048, 128, 0, stream>>>(x, fwp, X1);
  fno_dft_h<<<512, 256, 0, stream>>>(X1, tw, X2);
  fno_mix_idft_h<<<512, 256, 0, stream>>>(X2, wgt, tw, Y1);
  fno_idft_w<<<1024, 256, 0, stream>>>(Y1, cwp, out);
}